// FastEctLayer_73065983639654
// MI455X (gfx1250) — compile-verified
//
#include <hip/hip_runtime.h>

typedef __attribute__((ext_vector_type(2))) float v2f;
typedef __attribute__((ext_vector_type(8))) float v8f;

#define NTG 8  // theta groups of 16; WMMA kernel assumes T == 128

// ---------------------------------------------------------------------------
// Kernel 1: zero the histogram / output buffer (d_out is poisoned by harness).
// ---------------------------------------------------------------------------
__global__ void ect_zero_kernel(float* __restrict__ out, int n) {
    int i = blockIdx.x * blockDim.x + threadIdx.x;
    if (i < n) out[i] = 0.0f;
}

// ---------------------------------------------------------------------------
// Kernel 2: projection via V_WMMA_F32_16X16X4_F32 + histogram scatter.
//
// One wave owns a tile of 16 points x 128 thetas (8 WMMAs, K=4 with the 4th
// K slot zero-padded -> the K=3 dot product x . v[:,theta]).
//
// A layout (16x4 f32, 2 VGPRs): lanes 0-15 -> M=lane, K=0/1 in VGPR0/1;
//                               lanes 16-31 -> same M, K=2/3.
// B layout (4x16 f32, 2 VGPRs): mirrors A with N=lane&15.
// D layout (16x16 f32, 8 VGPRs): VGPR r, lane l -> M = r + (l<16?0:8),
//                               N = l&15.
// ---------------------------------------------------------------------------
__global__ __launch_bounds__(256) void ect_hist_wmma_kernel(
    const float* __restrict__ x,      // [N,3] row-major
    const float* __restrict__ v,      // [3,T] row-major, T == 128
    const int*   __restrict__ batch,  // [N], sorted
    const int*   __restrict__ res_ptr,// scalar resolution
    float*       __restrict__ hist,   // [B, res, T]
    int N, int T, int ntiles)
{
    const int   res   = res_ptr[0];
    const float scale = 0.5f * (float)res;
    const int   lane  = threadIdx.x & 31;
    const int   nlo   = lane & 15;
    const bool  lo    = lane < 16;
    const int   koff  = lo ? 0 : 2;     // which x component feeds K0

    const int wave   = (blockIdx.x * blockDim.x + threadIdx.x) >> 5;
    const int nwaves = (gridDim.x * blockDim.x) >> 5;

    // Preload B (v columns) for all 8 theta groups — loop-invariant.
    v2f Bm[NTG];
    #pragma unroll
    for (int tg = 0; tg < NTG; ++tg) {
        int n = tg * 16 + nlo;
        Bm[tg].x = lo ? v[n] : v[2 * T + n];
        Bm[tg].y = lo ? v[T + n] : 0.0f;
    }

    for (int tile = wave; tile < ntiles; tile += nwaves) {
        const int  base = tile * 16;
        const bool full = (base + 16 <= N);   // wave-uniform

        // ---- Build A: two unconditional b32 loads + one cndmask ----
        int pA = base + nlo;
        int pc = (pA < N) ? pA : (N - 1);
        int off = pc * 3 + koff;
        float ax = x[off];
        float ay = x[off + (lo ? 1 : 0)];     // hi lanes reload same dword
        v2f A;
        A.x = ax;
        A.y = lo ? ay : 0.0f;                 // K=3 pad for hi lanes
        if (!full) {                          // tail only (never taken here)
            bool valid = (pA < N);
            A.x = valid ? A.x : 0.0f;
            A.y = valid ? A.y : 0.0f;
        }

        // ---- 8 branch-free WMMAs: nh for 16 points x 128 thetas ----
        v8f acc[NTG];
        #pragma unroll
        for (int tg = 0; tg < NTG; ++tg) {
            v8f c = {};
            acc[tg] = __builtin_amdgcn_wmma_f32_16x16x4_f32(
                /*neg_a=*/false, A, /*neg_b=*/false, Bm[tg],
                /*c_mod=*/(short)0, c, /*reuse_a=*/false, /*reuse_b=*/false);
        }

        // One batch load per lane; redistribute per-row via shuffle:
        // lane l holds batch[base + (l&15)]; row r needs lane r (lo) / r+8 (hi).
        int bl = batch[pc];

        if (full) {
            #pragma unroll
            for (int r = 0; r < 8; ++r) {
                int b       = __shfl(bl, r + (lo ? 0 : 8));
                int rowbase = (b * res) * T;
                #pragma unroll
                for (int tg = 0; tg < NTG; ++tg) {
                    float nh  = acc[tg][r];
                    int   bin = (int)floorf((nh + 1.0f) * scale);
                    bin = bin < 0 ? 0 : (bin > res - 1 ? res - 1 : bin);
                    atomicAdd(&hist[rowbase + bin * T + tg * 16 + nlo], 1.0f);
                }
            }
        } else {
            #pragma unroll
            for (int r = 0; r < 8; ++r) {
                int b = __shfl(bl, r + (lo ? 0 : 8));
                int p = base + r + (lo ? 0 : 8);
                if (p < N) {
                    int rowbase = (b * res) * T;
                    #pragma unroll
                    for (int tg = 0; tg < NTG; ++tg) {
                        float nh  = acc[tg][r];
                        int   bin = (int)floorf((nh + 1.0f) * scale);
                        bin = bin < 0 ? 0 : (bin > res - 1 ? res - 1 : bin);
                        atomicAdd(&hist[rowbase + bin * T + tg * 16 + nlo], 1.0f);
                    }
                }
            }
        }
    }
}

// ---------------------------------------------------------------------------
// Generic fallback (any T): straightforward per-(point,theta) scatter.
// Only launched if T != 128.
// ---------------------------------------------------------------------------
__global__ void ect_hist_generic_kernel(
    const float* __restrict__ x, const float* __restrict__ v,
    const int* __restrict__ batch, const int* __restrict__ res_ptr,
    float* __restrict__ hist, int N, int T)
{
    int   res   = res_ptr[0];
    float scale = 0.5f * (float)res;
    int   tid   = blockIdx.x * blockDim.x + threadIdx.x;
    int   total = N * T;
    int   step  = gridDim.x * blockDim.x;
    for (int i = tid; i < total; i += step) {
        int p = i / T;
        int t = i - p * T;
        float nh = x[p*3+0] * v[t] + x[p*3+1] * v[T + t] + x[p*3+2] * v[2*T + t];
        int bin = (int)floorf((nh + 1.0f) * scale);
        bin = bin < 0 ? 0 : (bin > res - 1 ? res - 1 : bin);
        atomicAdd(&hist[(batch[p] * res + bin) * T + t], 1.0f);
    }
}

// ---------------------------------------------------------------------------
// Kernel 3: in-place cumsum along the filtration (resolution) axis.
// One thread per (batch, theta) column; consecutive threads -> consecutive
// theta -> coalesced along the fastest axis.
// ---------------------------------------------------------------------------
__global__ void ect_cumsum_kernel(float* __restrict__ hist,
                                  const int* __restrict__ res_ptr,
                                  int T, int out_size)
{
    int res  = res_ptr[0];
    int cols = out_size / res;           // B * T
    int tid  = blockIdx.x * blockDim.x + threadIdx.x;
    if (tid >= cols) return;
    int b = tid / T;
    int t = tid - b * T;
    long basei = (long)b * res * T + t;
    float run = 0.0f;
    for (int r = 0; r < res; ++r) {
        long idx = basei + (long)r * T;
        run += hist[idx];
        hist[idx] = run;
    }
}

extern "C" void kernel_launch(void* const* d_in, const int* in_sizes, int n_in,
                              void* d_out, int out_size, void* d_ws, size_t ws_size,
                              hipStream_t stream) {
    (void)n_in; (void)d_ws; (void)ws_size;
    const float* x       = (const float*)d_in[0];
    const float* v       = (const float*)d_in[1];
    const int*   batch   = (const int*)d_in[2];
    const int*   res_ptr = (const int*)d_in[3];
    // d_in[4] (batch_size) implied by out_size; unused.
    float* out = (float*)d_out;

    const int N = in_sizes[0] / 3;   // 200000
    const int T = in_sizes[1] / 3;   // 128

    // 1) zero histogram (d_out is poisoned).
    ect_zero_kernel<<<(out_size + 255) / 256, 256, 0, stream>>>(out, out_size);

    // 2) projection + atomic histogram.
    if (T == 16 * NTG) {
        const int ntiles = (N + 15) / 16;
        int blocks = 512;                       // 4096 waves, ~3 tiles each
        int maxBlocks = (ntiles + 7) / 8;       // 8 waves per 256-thread block
        if (blocks > maxBlocks) blocks = maxBlocks;
        if (blocks < 1) blocks = 1;
        ect_hist_wmma_kernel<<<blocks, 256, 0, stream>>>(x, v, batch, res_ptr,
                                                         out, N, T, ntiles);
    } else {
        int total = N * T;
        int blocks = (total + 255) / 256;
        if (blocks > 4096) blocks = 4096;
        ect_hist_generic_kernel<<<blocks, 256, 0, stream>>>(x, v, batch,
                                                            res_ptr, out, N, T);
    }

    // 3) cumsum along resolution axis, in place.
    ect_cumsum_kernel<<<(out_size + 255) / 256, 256, 0, stream>>>(out, res_ptr,
                                                                  T, out_size);
}